// Hyp_plus_MLR_70549132804742
// MI455X (gfx1250) — compile-verified
//
#include <hip/hip_runtime.h>
#include <math.h>

typedef __attribute__((ext_vector_type(2))) float v2f;
typedef __attribute__((ext_vector_type(8))) float v8f;
typedef __attribute__((__vector_size__(4 * sizeof(int)))) int i32x4;

#define B_SZ 2048
#define K_SZ 128
#define D_SZ 256
// LDS row stride in floats: 256 + 4 pad. Column reads (stride LDS_STRIDE) then
// land on banks {0,4,8,...,60} across the 16 lanes -> conflict-free.
#define LDS_STRIDE 260
#define WPB 4   // waves per block; each wave owns one 16-row M-tile x all K

#ifndef __has_builtin
#define __has_builtin(x) 0
#endif
#if __has_builtin(__builtin_amdgcn_global_load_async_to_lds_b128)
#define USE_ASYNC_LDS 1
#else
#define USE_ASYNC_LDS 0
#endif

#define AS1 __attribute__((address_space(1)))
#define AS3 __attribute__((address_space(3)))

// ---------------------------------------------------------------------------
// Per-k scalar precompute: x_k and a_hat_k are scalar multiples of z_k.
//   x_k     = scale_k * z_k,  scale_k = -(tanh(un)/un) * r_k,  un = max(|r|*|z|, eps)
//   a_hat_k = inva_k  * z_k,  inva_k  = sech(r)^2 / max(an, 1e-12), an = |z|*sech(r)^2
// ws layout (floats): [0:K) scale, [K:2K) inva, [2K:3K) x2, [3K:4K) xa, [4K:5K) an,
//                     [5K : 5K+B) y2
// ---------------------------------------------------------------------------
__global__ void hmlr_precompute_k(const float* __restrict__ z,
                                  const float* __restrict__ r,
                                  float* __restrict__ ws) {
  int k = threadIdx.x;
  if (k >= K_SZ) return;
  const float* zr = z + (size_t)k * D_SZ;
  float nz2 = 0.0f;
#pragma unroll 4
  for (int d = 0; d < D_SZ; d += 4) {
    float4 v = *(const float4*)(zr + d);
    nz2 += v.x * v.x + v.y * v.y + v.z * v.z + v.w * v.w;
  }
  float nz = sqrtf(nz2);
  float rv = r[k];
  float ch = coshf(rv);
  float sa = 1.0f / (ch * ch);
  float an = nz * sa;
  float un = fmaxf(fabsf(rv) * nz, 1e-15f);
  float t = tanhf(un);
  float scale = -(t / un) * rv;            // x = scale * z
  float inva = sa / fmaxf(an, 1e-12f);     // a_hat = inva * z
  ws[0 * K_SZ + k] = scale;
  ws[1 * K_SZ + k] = inva;
  ws[2 * K_SZ + k] = scale * scale * nz2;  // x2 = |x|^2
  ws[3 * K_SZ + k] = scale * inva * nz2;   // xa = dot(x, a_hat)
  ws[4 * K_SZ + k] = an;
}

__global__ void hmlr_precompute_y2(const float* __restrict__ y,
                                   float* __restrict__ y2out) {
  int b = blockIdx.x * blockDim.x + threadIdx.x;
  const float* yr = y + (size_t)b * D_SZ;
  float s = 0.0f;
#pragma unroll 4
  for (int d = 0; d < D_SZ; d += 4) {
    float4 v = *(const float4*)(yr + d);
    s += v.x * v.x + v.y * v.y + v.z * v.z + v.w * v.w;
  }
  y2out[b] = s;
}

// ---------------------------------------------------------------------------
// Fused GEMM (G = Y * Z^T via V_WMMA_F32_16X16X4_F32) + hyperbolic MLR epilogue.
// Block = WPB waves; wave w owns M-tile (blockIdx.x*WPB + w) x all 8 N-tiles
// (K=128). Full Z (128x256 f32) is staged into padded LDS per block via the
// CDNA5 async VGPR-bypassing path when available (ASYNCcnt-tracked).
// ---------------------------------------------------------------------------
__global__ void __launch_bounds__(WPB * 32)
hmlr_wmma_gemm(const float* __restrict__ y, const float* __restrict__ z,
               const float* __restrict__ ws, float* __restrict__ out) {
  extern __shared__ float zs[];

  const float* kscale = ws + 0 * K_SZ;
  const float* kinva  = ws + 1 * K_SZ;
  const float* kx2v   = ws + 2 * K_SZ;
  const float* kxav   = ws + 3 * K_SZ;
  const float* kanv   = ws + 4 * K_SZ;
  const float* y2v    = ws + 5 * K_SZ;

  // Cooperative stage of Z into LDS (b128 granularity, padded rows).
#if USE_ASYNC_LDS
  for (int i = threadIdx.x; i < (K_SZ * D_SZ) / 4; i += blockDim.x) {
    int k = i >> 6;              // i / (D_SZ/4)
    int d = (i & 63) << 2;       // (i % 64) * 4
    __builtin_amdgcn_global_load_async_to_lds_b128(
        (AS1 i32x4*)(z + (size_t)k * D_SZ + d),
        (AS3 i32x4*)(zs + k * LDS_STRIDE + d),
        /*offset=*/0, /*cpol=*/0);
  }
  asm volatile("s_wait_asynccnt 0x0" ::: "memory");  // drain before barrier
#else
  for (int i = threadIdx.x; i < (K_SZ * D_SZ) / 4; i += blockDim.x) {
    int k = i >> 6;
    int d = (i & 63) << 2;
    float4 v = *(const float4*)(z + (size_t)k * D_SZ + d);
    *(float4*)(&zs[k * LDS_STRIDE + d]) = v;
  }
#endif
  __syncthreads();

  const int lane = threadIdx.x & 31;
  const int wave = threadIdx.x >> 5;
  const int mtile = blockIdx.x * WPB + wave;
  const int mrow = lane & 15;   // A: M index; B/C/D: N index
  const int hi = lane >> 4;     // selects K pair (A/B) and M half (C/D)

  // A fragment source: y[mtile*16 + mrow][dc + 2*hi + {0,1}]
  const float* arow = y + (size_t)(mtile * 16 + mrow) * D_SZ + hi * 2;

  // B fragment source per N-tile: zs[nt*16 + mrow][dc + 2*hi + {0,1}]
  const float* bbase[8];
#pragma unroll
  for (int nt = 0; nt < 8; ++nt)
    bbase[nt] = &zs[(nt * 16 + mrow) * LDS_STRIDE + hi * 2];

  v8f acc[8] = {};

#pragma unroll 4
  for (int dc = 0; dc < D_SZ; dc += 4) {
    v2f a = *(const v2f*)(arow + dc);
#pragma unroll
    for (int nt = 0; nt < 8; ++nt) {
      v2f b = *(const v2f*)(bbase[nt] + dc);
      acc[nt] = __builtin_amdgcn_wmma_f32_16x16x4_f32(
          /*neg_a=*/false, a, /*neg_b=*/false, b,
          /*c_mod=*/(short)0, acc[nt], /*reuse_a=*/false, /*reuse_b=*/false);
    }
  }

  // Row-dependent scalars: this lane's 8 output rows (independent of N-tile).
  float y2r[8];
#pragma unroll
  for (int v = 0; v < 8; ++v) y2r[v] = y2v[mtile * 16 + v + 8 * hi];

  // Epilogue: closed-form hyperbolic MLR logit from G, y2_b and per-k scalars.
#pragma unroll
  for (int nt = 0; nt < 8; ++nt) {
    int k = nt * 16 + mrow;
    float ks = kscale[k];
    float kiv = kinva[k];
    float kx2 = kx2v[k];
    float kxa = kxav[k];
    float kan = kanv[k];
    float beta = 1.0f - kx2;
#pragma unroll
    for (int v = 0; v < 8; ++v) {
      int b = mtile * 16 + v + 8 * hi;
      float g = acc[nt][v];          // dot(y_b, z_k)
      float xy = ks * g;             // dot(x_k, y_b)
      float ya = kiv * g;            // dot(y_b, a_hat_k)
      float y2 = y2r[v];
      float alpha = 1.0f + 2.0f * xy + y2;
      float den = 1.0f + 2.0f * xy + kx2 * y2;
      float m2 = (alpha * alpha * kx2 + 2.0f * alpha * beta * xy +
                  beta * beta * y2) / (den * den);
      float lam = 2.0f / (1.0f - m2);
      float dotma = (alpha * kxa + beta * ya) / den;
      out[(size_t)b * K_SZ + k] = 2.0f * kan * asinhf(dotma * lam);
    }
  }
}

extern "C" void kernel_launch(void* const* d_in, const int* in_sizes, int n_in,
                              void* d_out, int out_size, void* d_ws, size_t ws_size,
                              hipStream_t stream) {
  const float* y = (const float*)d_in[0];   // output_before (2048, 256)
  const float* z = (const float*)d_in[1];   // z_mlr         (128, 256)
  const float* r = (const float*)d_in[2];   // mlr_r         (128, 1)
  float* ws = (float*)d_ws;                 // 5*K + B floats
  float* out = (float*)d_out;               // (2048, 128) row-major

  hmlr_precompute_k<<<1, K_SZ, 0, stream>>>(z, r, ws);
  hmlr_precompute_y2<<<B_SZ / 256, 256, 0, stream>>>(y, ws + 5 * K_SZ);

  size_t lds_bytes = (size_t)K_SZ * LDS_STRIDE * sizeof(float);  // 130 KB < 320 KB WGP LDS
  hmlr_wmma_gemm<<<B_SZ / (16 * WPB), WPB * 32, lds_bytes, stream>>>(y, z, ws, out);
}